// BasicGNN_25048249270599
// MI455X (gfx1250) — compile-verified
//
#include <hip/hip_runtime.h>
#include <math.h>

typedef __attribute__((ext_vector_type(2))) float v2f;
typedef __attribute__((ext_vector_type(8))) float v8f;

// ---------------------------------------------------------------------------
// Dense GEMM  H[M x 16] = act(A[M x K]) @ W[K x 16]  via V_WMMA_F32_16X16X4_F32
// One wave per 16-row tile. A-frag: lane half h holds K = k0+2h, k0+2h+1
// (contiguous float2). B-frag mirrors with N across lanes. C/D: vgpr r holds
// rows r (lanes 0-15) and r+8 (lanes 16-31).
// ---------------------------------------------------------------------------
template <int K, bool RELU>
__global__ void gemm_wmma_f32(const float* __restrict__ A,
                              const float* __restrict__ W,
                              float* __restrict__ H,
                              int num_tiles) {
    const int lane = threadIdx.x & 31;
    const int wave = threadIdx.x >> 5;
    const int tile = blockIdx.x * (blockDim.x >> 5) + wave;
    if (tile >= num_tiles) return;          // wave-uniform: EXEC stays all-1s

    const int half = lane >> 4;             // 0 or 1
    const int col  = lane & 15;             // M index for A, N index for B/D
    const int mrow = (tile << 4) + col;

    const float* arow = A + (size_t)mrow * K;
    v8f acc = {};

#pragma unroll
    for (int k0 = 0; k0 < K; k0 += 4) {
        const int k = k0 + (half << 1);
        v2f a, b;
        if (RELU) {
            a.x = fmaxf(arow[k], 0.0f);
            a.y = fmaxf(arow[k + 1], 0.0f);
        } else {
            a.x = arow[k];
            a.y = arow[k + 1];
        }
        b.x = W[k * 16 + col];
        b.y = W[(k + 1) * 16 + col];
        // 8 args: (neg_a, A, neg_b, B, c_mod, C, reuse_a, reuse_b)
        acc = __builtin_amdgcn_wmma_f32_16x16x4_f32(
            false, a, false, b, (short)0, acc, false, false);
    }

    float* hrow = H + ((size_t)(tile << 4) + (half << 3)) * 16 + col;
#pragma unroll
    for (int r = 0; r < 8; ++r) hrow[(size_t)r * 16] = acc[r];
}

// ---------------------------------------------------------------------------
// Degree / normalization helpers
// ---------------------------------------------------------------------------
__global__ void fill_ones(float* __restrict__ p, int n) {
    int i = blockIdx.x * blockDim.x + threadIdx.x;
    if (i < n) p[i] = 1.0f;                 // self-loop counted at dst
}

__global__ void deg_count(const int* __restrict__ dst, float* __restrict__ deg, int nE) {
    int e = blockIdx.x * blockDim.x + threadIdx.x;
    if (e < nE) atomicAdd(&deg[dst[e]], 1.0f);
}

__global__ void deg_rsqrt(const float* __restrict__ deg, float* __restrict__ dinv, int n) {
    int i = blockIdx.x * blockDim.x + threadIdx.x;
    if (i < n) dinv[i] = rsqrtf(deg[i]);    // deg >= 1 always (self-loops)
}

// out[i][c] = b[c] + dinv[i]^2 * h[i][c]   (bias + self-loop message)
__global__ void agg_init(const float* __restrict__ h, const float* __restrict__ dinv,
                         const float* __restrict__ bias, float* __restrict__ out,
                         int nNodes) {
    int gid = blockIdx.x * blockDim.x + threadIdx.x;
    if (gid >= nNodes * 16) return;
    int i = gid >> 4;
    int c = gid & 15;
    float di = dinv[i];
    out[gid] = bias[c] + di * di * h[gid];
}

// 16 lanes per edge: contiguous 64B gather of h[src], one f32 atomic per comp.
__global__ void edge_scatter(const int* __restrict__ src, const int* __restrict__ dst,
                             const float* __restrict__ dinv, const float* __restrict__ h,
                             float* __restrict__ out, int nE) {
    long long gid = (long long)blockIdx.x * blockDim.x + threadIdx.x;
    if (gid >= (long long)nE * 16) return;
    int e = (int)(gid >> 4);
    int c = (int)(gid & 15);
    int s = src[e];
    int d = dst[e];
    float val = h[(size_t)s * 16 + c] * dinv[s] * dinv[d];
    atomicAdd(&out[(size_t)d * 16 + c], val);
}

__global__ void zero_f32(float* __restrict__ p, int n) {
    int i = blockIdx.x * blockDim.x + threadIdx.x;
    if (i < n) p[i] = 0.0f;
}

// pooled[batch[i]][c] += relu(out[i][c])
__global__ void pool_relu(const float* __restrict__ out, const int* __restrict__ batch,
                          float* __restrict__ pooled, int nNodes) {
    int gid = blockIdx.x * blockDim.x + threadIdx.x;
    if (gid >= nNodes * 16) return;
    int i = gid >> 4;
    int c = gid & 15;
    float v = fmaxf(out[gid], 0.0f);
    atomicAdd(&pooled[(size_t)batch[i] * 16 + c], v);
}

__global__ void head_sigmoid(const float* __restrict__ pooled, const float* __restrict__ fcw,
                             const float* __restrict__ fcb, float* __restrict__ y, int G) {
    int g = blockIdx.x * blockDim.x + threadIdx.x;
    if (g >= G) return;
    float s = fcb[0];
#pragma unroll
    for (int n = 0; n < 16; ++n) s += pooled[(size_t)g * 16 + n] * fcw[n];
    y[g] = 1.0f / (1.0f + expf(-s));
}

// ---------------------------------------------------------------------------
extern "C" void kernel_launch(void* const* d_in, const int* in_sizes, int n_in,
                              void* d_out, int out_size, void* d_ws, size_t ws_size,
                              hipStream_t stream) {
    const float* x    = (const float*)d_in[0];   // [N,128]
    const float* W1   = (const float*)d_in[1];   // [128,16]
    const float* b1   = (const float*)d_in[2];   // [16]
    const float* W2   = (const float*)d_in[3];   // [16,16]
    const float* b2   = (const float*)d_in[4];   // [16]
    const float* fcw  = (const float*)d_in[5];   // [16,1]
    const float* fcb  = (const float*)d_in[6];   // [1]
    const int*   ei   = (const int*)d_in[7];     // [2,E]
    const int*   bat  = (const int*)d_in[8];     // [N]

    const int nNodes = in_sizes[8];          // 100000
    const int nE     = in_sizes[7] / 2;      // 3200000
    const int G      = out_size;             // 1024
    const int* srcA  = ei;
    const int* dstA  = ei + nE;

    // Workspace carve (aliased across layers): ~13.7 MB
    float* h      = (float*)d_ws;                    // [N*16] layer GEMM out
    float* out    = h    + (size_t)nNodes * 16;      // [N*16] aggregated out
    float* deg    = out  + (size_t)nNodes * 16;      // [N]
    float* dinv   = deg  + nNodes;                   // [N]
    float* pooled = dinv + nNodes;                   // [G*16]

    const int TB    = 256;                   // 8 waves / block (wave32)
    const int tiles = nNodes / 16;           // 6250, exact
    const int wpb   = TB / 32;
    const unsigned gTile = (tiles + wpb - 1) / wpb;
    const unsigned gNode = (unsigned)(((size_t)nNodes + TB - 1) / TB);
    const unsigned gFeat = (unsigned)(((size_t)nNodes * 16 + TB - 1) / TB);
    const unsigned gEdge = (unsigned)(((size_t)nE + TB - 1) / TB);
    const unsigned gEF   = (unsigned)(((size_t)nE * 16 + TB - 1) / TB);

    // Degrees + symmetric normalization
    fill_ones<<<gNode, TB, 0, stream>>>(deg, nNodes);
    deg_count<<<gEdge, TB, 0, stream>>>(dstA, deg, nE);
    deg_rsqrt<<<gNode, TB, 0, stream>>>(deg, dinv, nNodes);

    // Layer 1: h = x @ W1 (WMMA); out = b1 + dinv^2*h + scatter(norm * h[src])
    gemm_wmma_f32<128, false><<<gTile, TB, 0, stream>>>(x, W1, h, tiles);
    agg_init<<<gFeat, TB, 0, stream>>>(h, dinv, b1, out, nNodes);
    edge_scatter<<<gEF, TB, 0, stream>>>(srcA, dstA, dinv, h, out, nE);

    // Layer 2: h = relu(out) @ W2 (WMMA, relu fused into A-load); aggregate
    gemm_wmma_f32<16, true><<<gTile, TB, 0, stream>>>(out, W2, h, tiles);
    agg_init<<<gFeat, TB, 0, stream>>>(h, dinv, b2, out, nNodes);
    edge_scatter<<<gEF, TB, 0, stream>>>(srcA, dstA, dinv, h, out, nE);

    // Pool relu(out) by graph, then sigmoid head
    zero_f32<<<(G * 16 + TB - 1) / TB, TB, 0, stream>>>(pooled, G * 16);
    pool_relu<<<gFeat, TB, 0, stream>>>(out, bat, pooled, nNodes);
    head_sigmoid<<<(G + TB - 1) / TB, TB, 0, stream>>>(pooled, fcw, fcb, (float*)d_out, G);
}